// Net_18090402251166
// MI455X (gfx1250) — compile-verified
//
#include <hip/hip_runtime.h>

typedef __attribute__((ext_vector_type(2))) float v2f;
typedef __attribute__((ext_vector_type(8))) float v8f;

#define IN_DIM 128
#define HID 16

// ---------------- degree / normalization ----------------

__global__ void init_deg_kernel(float* __restrict__ deg, int n) {
    int i = blockIdx.x * blockDim.x + threadIdx.x;
    if (i < n) deg[i] = 1.0f;  // self-loop contributes 1 to in-degree
}

__global__ void count_deg_kernel(const long long* __restrict__ dst,
                                 float* __restrict__ deg, int e) {
    int i = blockIdx.x * blockDim.x + threadIdx.x;
    if (i < e) atomicAdd(&deg[(int)dst[i]], 1.0f);
}

__global__ void rsqrt_kernel(float* __restrict__ deg, int n) {
    int i = blockIdx.x * blockDim.x + threadIdx.x;
    if (i < n) deg[i] = rsqrtf(deg[i]);  // deg >= 1 always
}

// ---------------- WMMA GEMM: out[m][j] = dinv[m] * sum_k X[m][k]*W[k][j] ----
// One wave per 16-row tile. Uses V_WMMA_F32_16X16X4_F32 (full f32 precision).
// A layout (16x4 f32): lanes 0-15 hold rows M=0..15 with K=0,1 in vgpr0,1;
//                      lanes 16-31 hold the same rows with K=2,3.
// B layout (4x16 f32): lane holds column N=lane%16; lanes 0-15 hold K=0,1;
//                      lanes 16-31 hold K=2,3.
// C/D (16x16 f32): vgpr r = row (r + 8*half), lane%16 = column.

__global__ void gemm_scale_wmma(const float* __restrict__ X,
                                const float* __restrict__ W,
                                const float* __restrict__ dinv,
                                float* __restrict__ out,
                                int n, int K) {
    int wave    = (int)((blockIdx.x * blockDim.x + threadIdx.x) >> 5);
    int lane    = threadIdx.x & 31;
    int half    = lane >> 4;       // 0 or 1
    int l16     = lane & 15;
    int rowBase = wave * 16;

    int row = rowBase + l16;
    if (row > n - 1) row = n - 1;  // clamp (branch-free; EXEC stays all-1s)

    v8f acc = {};
    for (int k0 = 0; k0 < K; k0 += 4) {
        int ka = k0 + 2 * half;                      // even -> 8B aligned
        v2f a = *(const v2f*)(X + (long long)row * K + ka);
        v2f b;
        b.x = W[(ka + 0) * HID + l16];
        b.y = W[(ka + 1) * HID + l16];
        acc = __builtin_amdgcn_wmma_f32_16x16x4_f32(
            /*neg_a=*/false, a, /*neg_b=*/false, b,
            /*c_mod=*/(short)0, acc, /*reuse_a=*/false, /*reuse_b=*/false);
    }

#pragma unroll
    for (int r = 0; r < 8; ++r) {
        int m = rowBase + r + 8 * half;
        if (m < n) out[m * HID + l16] = acc[r] * dinv[m];  // dinv[m] uniform per 16 lanes
    }
}

// ---------------- aggregation ----------------

__global__ void copy_kernel(const float* __restrict__ src,
                            float* __restrict__ dst, int n) {
    int i = blockIdx.x * blockDim.x + threadIdx.x;
    if (i < n) dst[i] = src[i];
}

// thread t -> edge e = t/16, feature j = t%16.
// Lanes 0..15 read one coalesced 64B row of hs and hit 16 consecutive atomics.
__global__ void scatter_kernel(const long long* __restrict__ src,
                               const long long* __restrict__ dst,
                               const float* __restrict__ hs,
                               float* __restrict__ agg, int total) {
    int t = blockIdx.x * blockDim.x + threadIdx.x;
    if (t >= total) return;
    int e = t >> 4;
    int j = t & 15;
    int s = (int)src[e];
    int d = (int)dst[e];
    atomicAdd(&agg[d * HID + j], hs[s * HID + j]);
}

__global__ void finish_kernel(const float* __restrict__ agg,
                              const float* __restrict__ dinv,
                              const float* __restrict__ bias,
                              float* __restrict__ out, int n, int relu) {
    int t = blockIdx.x * blockDim.x + threadIdx.x;
    if (t >= n * HID) return;
    int i = t >> 4;
    int j = t & 15;
    float v = dinv[i] * agg[t] + bias[j];
    if (relu) v = fmaxf(v, 0.0f);
    out[t] = v;
}

// ---------------- driver ----------------

extern "C" void kernel_launch(void* const* d_in, const int* in_sizes, int n_in,
                              void* d_out, int out_size, void* d_ws, size_t ws_size,
                              hipStream_t stream) {
    const float*     x    = (const float*)d_in[0];
    const long long* ei   = (const long long*)d_in[1];  // int64 per reference
    const float*     W1   = (const float*)d_in[2];
    const float*     b1   = (const float*)d_in[3];
    const float*     W2   = (const float*)d_in[4];
    const float*     b2   = (const float*)d_in[5];
    float*           out  = (float*)d_out;

    const int N = in_sizes[0] / IN_DIM;   // 100000
    const int E = in_sizes[1] / 2;        // 3200000
    const long long* srcIdx = ei;
    const long long* dstIdx = ei + E;

    // workspace: dinv [N] | bufW [N*HID]   (~6.8 MB)
    float* dinv = (float*)d_ws;
    float* bufW = dinv + N;

    const int T = 256;
    dim3 blk(T);
    int gN  = (N + T - 1) / T;
    int gE  = (E + T - 1) / T;
    int gNF = (N * HID + T - 1) / T;
    int gEF = (E * HID + T - 1) / T;     // 51.2M threads, fits int
    int tiles = (N + 15) / 16;
    int gMM = (tiles * 32 + T - 1) / T;  // one wave per 16-row tile

    // normalization: deg -> dinv (in place)
    init_deg_kernel<<<gN, blk, 0, stream>>>(dinv, N);
    count_deg_kernel<<<gE, blk, 0, stream>>>(dstIdx, dinv, E);
    rsqrt_kernel<<<gN, blk, 0, stream>>>(dinv, N);

    // layer 1: hs1 = (x @ W1) * dinv  -> bufW
    gemm_scale_wmma<<<gMM, blk, 0, stream>>>(x, W1, dinv, bufW, N, IN_DIM);
    // agg1 (in d_out): self-loop init + edge scatter-add
    copy_kernel<<<gNF, blk, 0, stream>>>(bufW, out, N * HID);
    scatter_kernel<<<gEF, blk, 0, stream>>>(srcIdx, dstIdx, bufW, out, E * HID);
    // h1 = relu(dinv*agg1 + b1) -> bufW
    finish_kernel<<<gNF, blk, 0, stream>>>(out, dinv, b1, bufW, N, 1);

    // layer 2: hs2 = (h1 @ W2) * dinv -> d_out (agg1 dead)
    gemm_scale_wmma<<<gMM, blk, 0, stream>>>(bufW, W2, dinv, out, N, HID);
    // agg2 (in bufW): self-loop init + edge scatter-add
    copy_kernel<<<gNF, blk, 0, stream>>>(out, bufW, N * HID);
    scatter_kernel<<<gEF, blk, 0, stream>>>(srcIdx, dstIdx, out, bufW, E * HID);
    // final: out = dinv*agg2 + b2 (no relu)
    finish_kernel<<<gNF, blk, 0, stream>>>(bufW, dinv, b2, out, N, 0);
}